// VVCInterDCT2QuantIDCT2STEStable_51058571215569
// MI455X (gfx1250) — compile-verified
//
#include <hip/hip_runtime.h>

// CDNA5 / gfx1250: wave32, WMMA 16x16 shapes, f32 path = V_WMMA_F32_16X16X4_F32
typedef float v2f __attribute__((ext_vector_type(2)));
typedef float v8f __attribute__((ext_vector_type(8)));

#define QSTEP      25.5f
#define DZ_OFF     0.166015625f            // 85/512
#define Q_SCALE    40.1176470588235294f    // 1023/25.5
#define INV_MAXV   (1.0f/1023.0f)
#define CORR_SCALE (QSTEP/1024.0f)

// f32 WMMA 16x16x4: A is 16x4 (2 VGPRs/lane), B is 4x16 (2 VGPRs/lane), C/D 16x16 (8 VGPRs/lane)
#define WMMA_F32(a, b, c) \
    __builtin_amdgcn_wmma_f32_16x16x4_f32(false, (a), false, (b), (short)0, (c), false, false)

#define LDSTRIDE 33   // 32 + 1 pad to spread banks

__global__ __launch_bounds__(256)
void vvc_dct2_quant_idct2_kernel(const float* __restrict__ x,
                                 const float* __restrict__ Cmat,
                                 const float* __restrict__ cshift,
                                 float* __restrict__ out,
                                 int ntiles)
{
    __shared__ float ldsC[32 * LDSTRIDE];     // DCT matrix C, row-major, padded
    __shared__ float ldsCS[64];               // coeff_shift table
    __shared__ float buf[8][32 * LDSTRIDE];   // per-wave staging buffer

    const int tid  = threadIdx.x;
    const int lane = tid & 31;
    const int w    = tid >> 5;     // wave id within block (8 waves)
    const int lo   = lane & 15;    // M/N index within 16-block
    const int hi   = lane >> 4;    // selects K pair (A/B frags) / M half (C/D frags)

    // Broadcast C and coeff_shift into LDS (once per block)
    for (int i = tid; i < 1024; i += 256)
        ldsC[(i >> 5) * LDSTRIDE + (i & 31)] = Cmat[i];
    if (tid < 64) ldsCS[tid] = cshift[tid];
    __syncthreads();

    const int tile = blockIdx.x * 8 + w;
    if (tile >= ntiles) return;                 // whole-wave exit; EXEC stays all-1s for WMMA

    const float* xt = x   + (size_t)tile * 1024;
    float*       ot = out + (size_t)tile * 1024;
    float*       B  = buf[w];

    // ---- Load X tile (32x32 f32) into this wave's LDS buffer (coalesced float4 loads)
    #pragma unroll
    for (int i = 0; i < 8; ++i) {
        const float4 v = ((const float4*)xt)[i * 32 + lane];
        const int e = i * 128 + lane * 4;
        const int r = e >> 5, c = e & 31;
        float* p = &B[r * LDSTRIDE + c];
        p[0] = v.x; p[1] = v.y; p[2] = v.z; p[3] = v.w;
    }

    // Fragment loaders (A-frag lane map == B-frag lane map for f32 16x16x4):
    //   rowFrag: element [rbase+lo][kbase + 2*hi + j]  -> A-frag of M, B-frag of M^T
    //   colFrag: element [kbase + 2*hi + j][cbase+lo]  -> A-frag of M^T, B-frag of M
    #define ROWFRAG(M, rbase, kbase) \
        (v2f){ (M)[((rbase) + lo) * LDSTRIDE + (kbase) + 2 * hi], \
               (M)[((rbase) + lo) * LDSTRIDE + (kbase) + 2 * hi + 1] }
    #define COLFRAG(M, kbase, cbase) \
        (v2f){ (M)[((kbase) + 2 * hi) * LDSTRIDE + (cbase) + lo], \
               (M)[((kbase) + 2 * hi + 1) * LDSTRIDE + (cbase) + lo] }

    v8f a00, a01, a10, a11;

    // ---- GEMM 1: T = C @ X   (A = C row-read, B = X col-read)
    {
        v8f z = {};
        a00 = z; a01 = z; a10 = z; a11 = z;
        #pragma unroll
        for (int s = 0; s < 8; ++s) {
            const int kb = 4 * s;
            v2f aC0 = ROWFRAG(ldsC, 0,  kb);
            v2f aC1 = ROWFRAG(ldsC, 16, kb);
            v2f bX0 = COLFRAG(B, kb, 0);
            v2f bX1 = COLFRAG(B, kb, 16);
            a00 = WMMA_F32(aC0, bX0, a00);
            a01 = WMMA_F32(aC0, bX1, a01);
            a10 = WMMA_F32(aC1, bX0, a10);
            a11 = WMMA_F32(aC1, bX1, a11);
        }
    }

    // ---- Store T into LDS buffer (D-frag: row = 16*bi + v + 8*hi, col = 16*bj + lo)
    #pragma unroll
    for (int v = 0; v < 8; ++v) {
        const int r0 = v + 8 * hi;
        B[r0 * LDSTRIDE + lo]             = a00[v];
        B[r0 * LDSTRIDE + 16 + lo]        = a01[v];
        B[(16 + r0) * LDSTRIDE + lo]      = a10[v];
        B[(16 + r0) * LDSTRIDE + 16 + lo] = a11[v];
    }

    // ---- GEMM 2: coeff = T @ C^T   (A = T row-read, B = C row-read)
    {
        v8f z = {};
        v8f c00 = z, c01 = z, c10 = z, c11 = z;
        #pragma unroll
        for (int s = 0; s < 8; ++s) {
            const int kb = 4 * s;
            v2f aT0 = ROWFRAG(B, 0,  kb);
            v2f aT1 = ROWFRAG(B, 16, kb);
            v2f bC0 = ROWFRAG(ldsC, 0,  kb);
            v2f bC1 = ROWFRAG(ldsC, 16, kb);
            c00 = WMMA_F32(aT0, bC0, c00);
            c01 = WMMA_F32(aT0, bC1, c01);
            c10 = WMMA_F32(aT1, bC0, c10);
            c11 = WMMA_F32(aT1, bC1, c11);
        }
        a00 = c00; a01 = c01; a10 = c10; a11 = c11;
    }

    // ---- Pointwise dead-zone quantize + interpolated dequantize (forward of STE)
    // craw = coeff/1023;  qa = floor(|craw|*1023/QSTEP + DZ);
    // coeff_abs_hat = qa*QSTEP + (qa<64 ? cs[qa]*QSTEP/1024 : 0);  out = sgn*hat/1023
    #pragma unroll
    for (int blk = 0; blk < 4; ++blk) {
        v8f* acc = (blk == 0) ? &a00 : (blk == 1) ? &a01 : (blk == 2) ? &a10 : &a11;
        #pragma unroll
        for (int v = 0; v < 8; ++v) {
            const float craw = (*acc)[v];
            const float sgn  = __builtin_copysignf(1.0f, craw);
            const float qa   = floorf(fabsf(craw) * Q_SCALE + DZ_OFF);
            int idx = (int)qa; idx = idx > 63 ? 63 : idx;
            const float cs   = ldsCS[idx];
            const float cabs = qa * QSTEP;
            const float hat  = (qa < 64.0f) ? fmaf(cs, CORR_SCALE, cabs) : cabs;
            (*acc)[v] = hat * sgn * INV_MAXV;
        }
    }

    // ---- Store coeff_hat into LDS buffer
    #pragma unroll
    for (int v = 0; v < 8; ++v) {
        const int r0 = v + 8 * hi;
        B[r0 * LDSTRIDE + lo]             = a00[v];
        B[r0 * LDSTRIDE + 16 + lo]        = a01[v];
        B[(16 + r0) * LDSTRIDE + lo]      = a10[v];
        B[(16 + r0) * LDSTRIDE + 16 + lo] = a11[v];
    }

    // ---- GEMM 3: U = C^T @ H   (A = C col-read, B = H col-read)
    {
        v8f z = {};
        v8f u00 = z, u01 = z, u10 = z, u11 = z;
        #pragma unroll
        for (int s = 0; s < 8; ++s) {
            const int kb = 4 * s;
            v2f aC0 = COLFRAG(ldsC, kb, 0);
            v2f aC1 = COLFRAG(ldsC, kb, 16);
            v2f bH0 = COLFRAG(B, kb, 0);
            v2f bH1 = COLFRAG(B, kb, 16);
            u00 = WMMA_F32(aC0, bH0, u00);
            u01 = WMMA_F32(aC0, bH1, u01);
            u10 = WMMA_F32(aC1, bH0, u10);
            u11 = WMMA_F32(aC1, bH1, u11);
        }
        a00 = u00; a01 = u01; a10 = u10; a11 = u11;
    }

    // ---- Store U into LDS buffer
    #pragma unroll
    for (int v = 0; v < 8; ++v) {
        const int r0 = v + 8 * hi;
        B[r0 * LDSTRIDE + lo]             = a00[v];
        B[r0 * LDSTRIDE + 16 + lo]        = a01[v];
        B[(16 + r0) * LDSTRIDE + lo]      = a10[v];
        B[(16 + r0) * LDSTRIDE + 16 + lo] = a11[v];
    }

    // ---- GEMM 4: rec = U @ C   (A = U row-read, B = C col-read)
    {
        v8f z = {};
        v8f r00 = z, r01 = z, r10 = z, r11 = z;
        #pragma unroll
        for (int s = 0; s < 8; ++s) {
            const int kb = 4 * s;
            v2f aU0 = ROWFRAG(B, 0,  kb);
            v2f aU1 = ROWFRAG(B, 16, kb);
            v2f bC0 = COLFRAG(ldsC, kb, 0);
            v2f bC1 = COLFRAG(ldsC, kb, 16);
            r00 = WMMA_F32(aU0, bC0, r00);
            r01 = WMMA_F32(aU0, bC1, r01);
            r10 = WMMA_F32(aU1, bC0, r10);
            r11 = WMMA_F32(aU1, bC1, r11);
        }
        a00 = r00; a01 = r01; a10 = r10; a11 = r11;
    }

    // ---- Store result straight from D-fragments (16 contiguous floats / half-wave)
    #pragma unroll
    for (int v = 0; v < 8; ++v) {
        const int r0 = v + 8 * hi;
        ot[r0 * 32 + lo]             = a00[v];
        ot[r0 * 32 + 16 + lo]        = a01[v];
        ot[(16 + r0) * 32 + lo]      = a10[v];
        ot[(16 + r0) * 32 + 16 + lo] = a11[v];
    }
}

extern "C" void kernel_launch(void* const* d_in, const int* in_sizes, int n_in,
                              void* d_out, int out_size, void* d_ws, size_t ws_size,
                              hipStream_t stream) {
    const float* x  = (const float*)d_in[0];   // residual_norm [256,64,32,32]
    const float* C  = (const float*)d_in[1];   // dct_mat [32,32]
    const float* cs = (const float*)d_in[2];   // coeff_shift [64]
    float* out = (float*)d_out;

    const int ntiles = in_sizes[0] / 1024;     // 16384 independent 32x32 tiles
    const int blocks = (ntiles + 7) / 8;       // 8 waves/block, 1 tile/wave

    hipLaunchKernelGGL(vvc_dct2_quant_idct2_kernel,
                       dim3(blocks), dim3(256), 0, stream,
                       x, C, cs, out, ntiles);
}